// VanillaGNN_87050397155999
// MI455X (gfx1250) — compile-verified
//
#include <hip/hip_runtime.h>
#include <hip/hip_bf16.h>
#include <math.h>

// ---------------------------------------------------------------------------
// VanillaGNN on MI455X (gfx1250, wave32).
//   H1 = X @ W1                     (WMMA f32 16x16x4; W1 K-panels double-
//                                    buffered in LDS via
//                                    global_load_async_to_lds_b128 / ASYNCcnt)
//   A1[i] = sum_{e: dst=i} H1[src]  (global_atomic_add_f32, L2-resident)
//   H2 = relu(A1) @ W2pad           (WMMA f32; W2 padded 64x40->64x56 in LDS)
//   OUT[i] = sum_{e} H2[src]        (atomics into d_out)
//   OUT = log_softmax(OUT, axis=1)  (in-place, wave-per-row, shfl_xor wave32)
// ---------------------------------------------------------------------------

typedef float v2f __attribute__((ext_vector_type(2)));
typedef float v8f __attribute__((ext_vector_type(8)));

#define N_NODES 50000
#define N_EDGES 800000
#define IN_DIM  512
#define HID_DIM 64
#define OUT_DIM 40
#define OUT_PAD 48        // 3 x 16 column tiles
#define G_WAVES 5         // 3125 row-strips = 625 blocks x 5 waves, exact
#define G_THREADS (G_WAVES * 32)
#define W1_STRIDE 72      // LDS row stride (floats): +8 banks/row -> conflict-free
#define W2_STRIDE 56      // LDS row stride (floats) for the 48-wide padded W2
#define N_PANELS (IN_DIM / 64)

// --------------------------------------------------- CDNA5 async LDS helpers
// Copy 16 bytes global -> LDS through the async path (tracked by ASYNCcnt).
// Low 32 bits of a flat LDS pointer are the LDS byte offset (aperture rule).
__device__ __forceinline__ void async_copy_b128(const float* gsrc, float* ldst) {
    unsigned ldsoff = (unsigned)(uintptr_t)ldst;
    asm volatile("global_load_async_to_lds_b128 %0, %1, off"
                 :: "v"(ldsoff), "v"(gsrc) : "memory");
}
__device__ __forceinline__ void wait_async_lds() {
    asm volatile("s_wait_asynccnt 0x0" ::: "memory");
}

// ---------------------------------------------------------------- utilities
__global__ __launch_bounds__(256) void zero_f32(float* __restrict__ p, int n) {
    int i = blockIdx.x * 256 + threadIdx.x;
    if (i < n) p[i] = 0.0f;
}

// ------------------------------------------------------------------- GEMM 1
// Block: 5 waves, each owning one 16-row strip (rows m0..m0+15), all 64 cols.
// W1 is consumed in 64-deep K-panels, DOUBLE-BUFFERED in LDS: panel p+1 is
// async-fetched while panel p is computed; s_wait_asynccnt 0 + barrier at the
// end of each panel. Buffer (p+1)&1 was last read in panel p-1, whose barrier
// already retired all reads, so the overlapped fill is hazard-free.
// A-frag (f32 16x4): lane<16 holds K=k,k+1 ; lane>=16 holds K=k+2,k+3 of row lane&15.
// B-frag mirrored from LDS (stride 72 -> bank-conflict-free for all 4 tiles).
// C/D: VGPR j -> row m0 + j + 8*half, col lane&15.
__global__ __launch_bounds__(G_THREADS) void gemm_x_w1(const float* __restrict__ X,
                                                       const float* __restrict__ W1,
                                                       float* __restrict__ H1) {
    __shared__ float ldsW[2][64 * W1_STRIDE];         // 2 x 18,432 B panels
    const int tid  = threadIdx.x;
    const int wave = tid >> 5;
    const int lane = tid & 31;
    const int half = lane >> 4;
    const int l15  = lane & 15;
    const int m0   = blockIdx.x * (16 * G_WAVES) + wave * 16;

    auto fill = [&](int buf, int k0) {
        // stage W1[k0..k0+63][0..63] -> LDS, 16B chunks: 64 rows x 16 chunks
        for (int c = tid; c < 64 * 16; c += G_THREADS) {
            int row = c >> 4, pos = c & 15;
            async_copy_b128(W1 + (size_t)(k0 + row) * HID_DIM + pos * 4,
                            &ldsW[buf][row * W1_STRIDE + pos * 4]);
        }
    };

    v8f c0 = {}, c1 = {}, c2 = {}, c3 = {};
    const float* xrow = X + (size_t)(m0 + l15) * IN_DIM;

    fill(0, 0);
    wait_async_lds();
    __syncthreads();

    for (int p = 0; p < N_PANELS; ++p) {
        if (p + 1 < N_PANELS) fill((p + 1) & 1, (p + 1) * 64);   // prefetch next

        const float* Wp = ldsW[p & 1];
        const float* xk = xrow + p * 64;
        for (int kk = 0; kk < 64; kk += 4) {
            v2f a = *(const v2f*)(xk + kk + 2 * half);           // contiguous pair
            const float* w0 = &Wp[(kk + 2 * half) * W1_STRIDE];
            const float* w1 = w0 + W1_STRIDE;
            v2f b0 = { w0[l15],      w1[l15]      };
            v2f b1 = { w0[16 + l15], w1[16 + l15] };
            v2f b2 = { w0[32 + l15], w1[32 + l15] };
            v2f b3 = { w0[48 + l15], w1[48 + l15] };
            c0 = __builtin_amdgcn_wmma_f32_16x16x4_f32(false, a, false, b0, (short)0, c0, false, false);
            c1 = __builtin_amdgcn_wmma_f32_16x16x4_f32(false, a, false, b1, (short)0, c1, false, false);
            c2 = __builtin_amdgcn_wmma_f32_16x16x4_f32(false, a, false, b2, (short)0, c2, false, false);
            c3 = __builtin_amdgcn_wmma_f32_16x16x4_f32(false, a, false, b3, (short)0, c3, false, false);
        }

        wait_async_lds();                             // next panel landed
        __syncthreads();                              // all waves done with p
    }

#pragma unroll
    for (int j = 0; j < 8; ++j) {
        float* o = H1 + (size_t)(m0 + j + 8 * half) * HID_DIM + l15;
        o[0]  = c0[j];
        o[16] = c1[j];
        o[32] = c2[j];
        o[48] = c3[j];
    }
}

// ------------------------------------------------------------------- GEMM 2
// H2[N x 48] = relu(A1[N x 64]) @ W2pad[64 x 48]; W2 (64x40) staged into a
// single 64 x 56-stride LDS panel with columns 40..47 zeroed (the pad cols).
__global__ __launch_bounds__(G_THREADS) void gemm_h_w2(const float* __restrict__ A1,
                                                       const float* __restrict__ W2,
                                                       float* __restrict__ H2) {
    __shared__ float ldsW[64 * W2_STRIDE];            // 14,336 B
    const int tid  = threadIdx.x;
    const int wave = tid >> 5;
    const int lane = tid & 31;
    const int half = lane >> 4;
    const int l15  = lane & 15;
    const int m0   = blockIdx.x * (16 * G_WAVES) + wave * 16;

    // stage W2 (40 floats = 10 x 16B chunks per row) + zero pad cols 40..47
    for (int c = tid; c < 64 * 10; c += G_THREADS) {
        int row = c / 10, pos = c - row * 10;
        async_copy_b128(W2 + (size_t)row * OUT_DIM + pos * 4,
                        &ldsW[row * W2_STRIDE + pos * 4]);
    }
    for (int i = tid; i < 64 * 8; i += G_THREADS) {
        int row = i >> 3, col = OUT_DIM + (i & 7);
        ldsW[row * W2_STRIDE + col] = 0.0f;
    }
    wait_async_lds();
    __syncthreads();

    v8f c0 = {}, c1 = {}, c2 = {};
    const float* arow = A1 + (size_t)(m0 + l15) * HID_DIM;

    for (int k = 0; k < HID_DIM; k += 4) {
        v2f a;
        a.x = fmaxf(arow[k + 2 * half],     0.0f);    // fused relu
        a.y = fmaxf(arow[k + 2 * half + 1], 0.0f);
        const float* w0 = &ldsW[(k + 2 * half) * W2_STRIDE];
        const float* w1 = w0 + W2_STRIDE;
        v2f b0 = { w0[l15],      w1[l15]      };
        v2f b1 = { w0[16 + l15], w1[16 + l15] };
        v2f b2 = { w0[32 + l15], w1[32 + l15] };
        c0 = __builtin_amdgcn_wmma_f32_16x16x4_f32(false, a, false, b0, (short)0, c0, false, false);
        c1 = __builtin_amdgcn_wmma_f32_16x16x4_f32(false, a, false, b1, (short)0, c1, false, false);
        c2 = __builtin_amdgcn_wmma_f32_16x16x4_f32(false, a, false, b2, (short)0, c2, false, false);
    }

#pragma unroll
    for (int j = 0; j < 8; ++j) {
        float* o = H2 + (size_t)(m0 + j + 8 * half) * OUT_PAD + l15;
        o[0]  = c0[j];
        o[16] = c1[j];
        o[32] = c2[j];
    }
}

// --------------------------------------------------------------- scatters
__global__ __launch_bounds__(256) void scatter64(const int* __restrict__ src,
                                                 const int* __restrict__ dst,
                                                 const float* __restrict__ H,
                                                 float* __restrict__ A) {
    int tid = blockIdx.x * 256 + threadIdx.x;     // grid exactly N_EDGES*64
    int e = tid >> 6, f = tid & 63;
    int s = src[e], d = dst[e];
    atomicAdd(&A[(size_t)d * HID_DIM + f], H[(size_t)s * HID_DIM + f]);
}

__global__ __launch_bounds__(256) void scatter40(const int* __restrict__ src,
                                                 const int* __restrict__ dst,
                                                 const float* __restrict__ H2,
                                                 float* __restrict__ out) {
    int tid = blockIdx.x * 256 + threadIdx.x;
    int e = tid >> 6, f = tid & 63;
    if (f < OUT_DIM) {
        int s = src[e], d = dst[e];
        atomicAdd(&out[(size_t)d * OUT_DIM + f], H2[(size_t)s * OUT_PAD + f]);
    }
}

// ---------------------------------------------------------- log_softmax(40)
__global__ __launch_bounds__(256) void logsoftmax40(float* __restrict__ out) {
    int wave = threadIdx.x >> 5;
    int lane = threadIdx.x & 31;
    int row  = blockIdx.x * 8 + wave;             // 6250 * 8 = 50000, exact
    float* r = out + (size_t)row * OUT_DIM;

    float v0 = r[lane];
    float v1 = (lane < 8) ? r[32 + lane] : -INFINITY;

    float m = fmaxf(v0, v1);
#pragma unroll
    for (int off = 16; off > 0; off >>= 1)
        m = fmaxf(m, __shfl_xor(m, off, 32));

    float s = expf(v0 - m) + ((lane < 8) ? expf(v1 - m) : 0.0f);
#pragma unroll
    for (int off = 16; off > 0; off >>= 1)
        s += __shfl_xor(s, off, 32);

    float l = logf(s);
    r[lane] = v0 - m - l;
    if (lane < 8) r[32 + lane] = v1 - m - l;
}

// ------------------------------------------------------------------ launch
extern "C" void kernel_launch(void* const* d_in, const int* in_sizes, int n_in,
                              void* d_out, int out_size, void* d_ws, size_t ws_size,
                              hipStream_t stream) {
    const float* X   = (const float*)d_in[0];
    const int*   src = (const int*)  d_in[1];
    const int*   dst = (const int*)  d_in[2];
    const float* W1  = (const float*)d_in[3];
    const float* W2  = (const float*)d_in[4];
    float* out = (float*)d_out;

    // workspace layout (floats): H1 [50000*64] | A1 [50000*64]
    float* H1 = (float*)d_ws;
    float* A1 = H1 + (size_t)N_NODES * HID_DIM;
    float* H2 = H1;   // H1 dead after scatter64; 48-wide H2 fits its 64-wide slot

    // 1. zero accumulation buffers
    zero_f32<<<(N_NODES * HID_DIM + 255) / 256, 256, 0, stream>>>(A1, N_NODES * HID_DIM);
    zero_f32<<<(N_NODES * OUT_DIM + 255) / 256, 256, 0, stream>>>(out, N_NODES * OUT_DIM);

    // 2. H1 = X @ W1        (625 blocks x 5 waves, one 16-row strip per wave)
    gemm_x_w1<<<N_NODES / (16 * G_WAVES), G_THREADS, 0, stream>>>(X, W1, H1);

    // 3. A1 = segment_sum(H1[src], dst)
    scatter64<<<(N_EDGES * 64) / 256, 256, 0, stream>>>(src, dst, H1, A1);

    // 4. H2 = relu(A1) @ W2pad
    gemm_h_w2<<<N_NODES / (16 * G_WAVES), G_THREADS, 0, stream>>>(A1, W2, H2);

    // 5. out = segment_sum(H2[src], dst)
    scatter40<<<(N_EDGES * 64) / 256, 256, 0, stream>>>(src, dst, H2, out);

    // 6. out = log_softmax(out, axis=1)
    logsoftmax40<<<N_NODES / 8, 256, 0, stream>>>(out);
}